// SuperPixelMeanEmbed_54597624267482
// MI455X (gfx1250) — compile-verified
//
#include <hip/hip_runtime.h>
#include <hip/hip_bf16.h>

#define NSEG 196
#define COUT 56
#define HW   50176      // 224*224
#define NBATCH 32

typedef __attribute__((ext_vector_type(2))) float v2f;
typedef __attribute__((ext_vector_type(8))) float v8f;

// ---------------------------------------------------------------------------
// Kernel 0: zero the workspace accumulator bins (B x NSEG x 4 floats)
// ---------------------------------------------------------------------------
__global__ void spme_zero_ws(float* __restrict__ ws, int n) {
    int i = blockIdx.x * blockDim.x + threadIdx.x;
    if (i < n) ws[i] = 0.0f;
}

// ---------------------------------------------------------------------------
// Kernel 1: per-batch segment binning of the RAW 3-channel input + counts.
// Exploits linearity of the 1x1 conv: mean_seg(Wx+b) = W*mean_seg(x)+b, so we
// scatter only 4 floats/pixel instead of 56. LDS-privatized bins (ds_add_f32)
// then one global atomic per (block, bin).
// Grid: (49 chunks of 1024 px, NBATCH). Block: 256 threads, 4 consecutive
// pixels per thread -> all channel/seg loads are global_load_b128.
// ---------------------------------------------------------------------------
__global__ __launch_bounds__(256) void spme_bin_kernel(
    const float* __restrict__ X,       // (B, 3, HW) planar
    const int*   __restrict__ seg,     // (B, HW)
    float*       __restrict__ bins)    // (B, NSEG, 4): sumR,sumG,sumB,count
{
    __shared__ float lbin[NSEG * 4];
    const int b     = blockIdx.y;
    const int chunk = blockIdx.x;
    const int tid   = threadIdx.x;

    for (int i = tid; i < NSEG * 4; i += 256) lbin[i] = 0.0f;
    __syncthreads();

    const float4* Xb4   = (const float4*)(X   + (size_t)b * 3 * HW);
    const int4*   segb4 = (const int4*)  (seg + (size_t)b * HW);
    const int     q     = chunk * 256 + tid;   // index in float4 units

    int4   s4 = segb4[q];
    float4 r4 = Xb4[q];                        // channel 0
    float4 g4 = Xb4[q + HW / 4];               // channel 1
    float4 b4 = Xb4[q + 2 * (HW / 4)];         // channel 2

    const int   si[4] = { s4.x, s4.y, s4.z, s4.w };
    const float rv[4] = { r4.x, r4.y, r4.z, r4.w };
    const float gv[4] = { g4.x, g4.y, g4.z, g4.w };
    const float bv[4] = { b4.x, b4.y, b4.z, b4.w };

#pragma unroll
    for (int it = 0; it < 4; ++it) {
        int base = si[it] * 4;
        atomicAdd(&lbin[base + 0], rv[it]);    // ds_add_f32
        atomicAdd(&lbin[base + 1], gv[it]);
        atomicAdd(&lbin[base + 2], bv[it]);
        atomicAdd(&lbin[base + 3], 1.0f);
    }
    __syncthreads();

    float* gb = bins + (size_t)b * NSEG * 4;
    for (int i = tid; i < NSEG * 4; i += 256) {
        float v = lbin[i];
        if (v != 0.0f) atomicAdd(&gb[i], v);   // global_atomic_add_f32
    }
}

// ---------------------------------------------------------------------------
// Kernel 2: out[b,s,o] = sum_c W[o,c] * (binsum[b,s,c]/max(count,1)) + bias[o]
// via V_WMMA_F32_16X16X4_F32. One wave per 16-segment x 16-channel tile.
// Grid: (4 chan-tiles, 13 seg-tiles, NBATCH). Block: 32 (one wave).
//
// ISA layouts (wave32):
//   A 16x4 (M=seg,K=chan): A[m=lane&15][k=2*(lane>>4)+v], v = VGPR 0/1
//   B 4x16 (K=chan,N=out): B[k=2*(lane>>4)+v][n=lane&15]
//   D 16x16: D[m=j+8*(lane>>4)][n=lane&15], j = VGPR 0..7
// ---------------------------------------------------------------------------
__global__ __launch_bounds__(32) void spme_project_wmma(
    const float* __restrict__ bins,   // (B, NSEG, 4)
    const float* __restrict__ W,      // (COUT, 3) row-major
    const float* __restrict__ bias,   // (COUT)
    float*       __restrict__ out)    // (B, NSEG, COUT)
{
    const int colTile = blockIdx.x;           // 0..3   (channels 0..63, 56 used)
    const int rowTile = blockIdx.y;           // 0..12  (segments 0..207, 196 used)
    const int b       = blockIdx.z;
    const int lane    = threadIdx.x;
    const int hi      = lane >> 4;
    const int lo      = lane & 15;
    const int k0      = hi * 2;

    // A-matrix fragment: segment means (K padded 3 -> 4 with zeros)
    int s = rowTile * 16 + lo;
    v2f a; a.x = 0.0f; a.y = 0.0f;
    if (s < NSEG) {
        const float* bn = bins + ((size_t)b * NSEG + s) * 4;
        float inv = 1.0f / fmaxf(bn[3], 1.0f);
        if (k0     < 3) a.x = bn[k0]     * inv;
        if (k0 + 1 < 3) a.y = bn[k0 + 1] * inv;
    }

    // B-matrix fragment: W^T tile, B[k][n] = W[n*3 + k]
    int o = colTile * 16 + lo;
    v2f bm; bm.x = 0.0f; bm.y = 0.0f;
    if (o < COUT) {
        if (k0     < 3) bm.x = W[o * 3 + k0];
        if (k0 + 1 < 3) bm.y = W[o * 3 + k0 + 1];
    }

    v8f c = {};
    // 8 args: (neg_a, A, neg_b, B, c_mod, C, reuse_a, reuse_b)
    c = __builtin_amdgcn_wmma_f32_16x16x4_f32(
            false, a, false, bm, (short)0, c, false, false);

    if (o < COUT) {
        float bo = bias[o];
        const int m0 = rowTile * 16 + 8 * hi;           // first row of my half
        float* op = out + ((size_t)b * NSEG + m0) * COUT + o;
        if (m0 + 7 < NSEG) {
            // Common case: whole 8-row half in range, unguarded stores.
#pragma unroll
            for (int j = 0; j < 8; ++j)
                op[(size_t)j * COUT] = c[j] + bo;
        } else {
            // Tail tile (rowTile == 12): per-row guard.
#pragma unroll
            for (int j = 0; j < 8; ++j)
                if (m0 + j < NSEG)
                    op[(size_t)j * COUT] = c[j] + bo;
        }
    }
}

// ---------------------------------------------------------------------------
extern "C" void kernel_launch(void* const* d_in, const int* in_sizes, int n_in,
                              void* d_out, int out_size, void* d_ws, size_t ws_size,
                              hipStream_t stream) {
    const float* X    = (const float*)d_in[0];   // (32,3,224,224) fp32
    const int*   seg  = (const int*)  d_in[1];   // (32,224,224)   int32
    const float* W    = (const float*)d_in[2];   // (56,3)         fp32
    const float* bias = (const float*)d_in[3];   // (56,)          fp32
    float*       out  = (float*)d_out;           // (32,196,56)    fp32
    float*       bins = (float*)d_ws;            // 32*196*4 floats = 100 KB

    const int nbins = NBATCH * NSEG * 4;
    spme_zero_ws<<<(nbins + 255) / 256, 256, 0, stream>>>(bins, nbins);
    spme_bin_kernel<<<dim3(49, NBATCH), 256, 0, stream>>>(X, seg, bins);
    spme_project_wmma<<<dim3(4, 13, NBATCH), 32, 0, stream>>>(bins, W, bias, out);
}